// IBNet_86681029968319
// MI455X (gfx1250) — compile-verified
//
#include <hip/hip_runtime.h>
#include <math.h>

#define NS 1024   // samples
#define DF 64     // feature dim
#define NL 3      // IB layers
#define TI 16     // samples per pairwise block

typedef __attribute__((ext_vector_type(2))) float v2f;
typedef __attribute__((ext_vector_type(8))) float v8f;

__global__ void ib_zero(float* out) {
    if (threadIdx.x == 0) out[0] = 0.0f;
}

// One wave (32 lanes) computes one 16x16 tile of BOTH mu and var for one layer.
// mu[n,d]  = sum_k f_s[n,k] * W_mu[d,k]  + b_mu[d]      (i.e. f_s @ W.T)
// var[n,d] = softplus(f_s @ W_var.T + b_var)*0.1 + 1e-6
// Tiles: NL * (NS/16) * (DF/16) = 3 * 64 * 4 = 768, 4 waves per block -> 192 blocks.
__global__ __launch_bounds__(128) void ib_head_wmma(
    const float* __restrict__ fea1, const float* __restrict__ fea2,
    const float* __restrict__ fea3,
    const float* __restrict__ W_mu, const float* __restrict__ b_mu,
    const float* __restrict__ W_var, const float* __restrict__ b_var,
    float* __restrict__ mu_out, float* __restrict__ var_out)
{
    const int lane = threadIdx.x & 31;
    const int tile = blockIdx.x * (blockDim.x >> 5) + (threadIdx.x >> 5);

    const int layer = tile >> 8;       // tile / 256
    const int rem   = tile & 255;
    const int tm    = rem >> 2;        // 0..63 : 16-row tile of samples
    const int tn    = rem & 3;         // 0..3  : 16-col tile of features

    const float* f_s = (layer == 0) ? fea1 : (layer == 1) ? fea2 : fea3;
    const float* Wm  = W_mu  + layer * DF * DF;
    const float* Wv  = W_var + layer * DF * DF;

    // ISA 16x4 fp32 A layout: lanes 0-15 rows M, VGPR0 holds K=k0 (half 0) /
    // K=k0+2 (half 1), VGPR1 holds K=k0+1 / k0+3. B (4x16) mirrors with cols.
    const int half = lane >> 4;
    const int l    = lane & 15;
    const int row  = tm * 16 + l;      // A-matrix row (sample)
    const int col  = tn * 16 + l;      // B-matrix col (output feature d)

    v8f cmu  = {};
    v8f cvar = {};
    #pragma unroll
    for (int k0 = 0; k0 < DF; k0 += 4) {
        const int k = k0 + half * 2;
        v2f a, bm, bv;
        a.x  = f_s[row * DF + k];     a.y  = f_s[row * DF + k + 1];
        bm.x = Wm [col * DF + k];     bm.y = Wm [col * DF + k + 1];   // B[k,col] = W[col,k]
        bv.x = Wv [col * DF + k];     bv.y = Wv [col * DF + k + 1];
        cmu  = __builtin_amdgcn_wmma_f32_16x16x4_f32(false, a, false, bm,
                                                     (short)0, cmu,  false, false);
        cvar = __builtin_amdgcn_wmma_f32_16x16x4_f32(false, a, false, bv,
                                                     (short)0, cvar, false, false);
    }

    // C/D layout: VGPR v -> row (v + 8*half), col = lane&15.
    const float biasm = b_mu [layer * DF + col];
    const float biasv = b_var[layer * DF + col];
    float* mu_l  = mu_out  + (size_t)layer * NS * DF;
    float* var_l = var_out + (size_t)layer * NS * DF;
    #pragma unroll
    for (int v = 0; v < 8; ++v) {
        const int m   = tm * 16 + half * 8 + v;
        const float muv = cmu[v] + biasm;
        const float x   = cvar[v] + biasv;
        // stable softplus: max(x,0) + log1p(exp(-|x|))
        const float sp  = fmaxf(x, 0.0f) + log1pf(expf(-fabsf(x)));
        mu_l [m * DF + col] = muv;
        var_l[m * DF + col] = sp * 0.1f + 1e-6f;
    }
}

// One block per (layer, 16-sample tile). Stages f_r through LDS in 64-row
// chunks so each loaded element is reused TI=16 times. Computes
//   sum_d ( mean_j |f_r[j,d]-mu[i,d]| - |mu[i,d]-f_r[i,d]| ) / var[i,d]
// summed over the tile's samples, atomically accumulated /NS into out.
__global__ __launch_bounds__(256) void ib_pairwise(
    const float* __restrict__ fea0, const float* __restrict__ fea1,
    const float* __restrict__ fea2,
    const float* __restrict__ mu_ws, const float* __restrict__ var_ws,
    float* __restrict__ out)
{
    __shared__ float fr[64 * DF];   // 16 KB chunk of f_r
    __shared__ float red[256];

    const int layer = blockIdx.y;
    const int i0    = blockIdx.x * TI;
    const int t     = threadIdx.x;
    const int d     = t & (DF - 1);   // feature dim owned by this thread
    const int irow  = t >> 6;         // 0..3 : base sample-row within tile

    const float* f_r = (layer == 0) ? fea0 : (layer == 1) ? fea1 : fea2;
    const size_t base = (size_t)layer * NS * DF;

    // this thread owns samples i0 + irow + 4*p, p = 0..3, at feature d
    float muv[4], acc[4];
    #pragma unroll
    for (int p = 0; p < 4; ++p) {
        muv[p] = mu_ws[base + (size_t)(i0 + irow + 4 * p) * DF + d];
        acc[p] = 0.0f;
    }

    for (int c = 0; c < NS / 64; ++c) {
        __syncthreads();
        // cooperative load: 64 rows x 64 floats = 1024 float4s, 4 per thread
        const float4* src = (const float4*)(f_r + (size_t)c * 64 * DF);
        float4*       dst = (float4*)fr;
        #pragma unroll
        for (int q = 0; q < 4; ++q)
            dst[t + q * 256] = src[t + q * 256];
        __syncthreads();

        #pragma unroll 8
        for (int j = 0; j < 64; ++j) {
            const float fv = fr[j * DF + d];   // lane->bank = d: conflict-free
            acc[0] += fabsf(fv - muv[0]);
            acc[1] += fabsf(fv - muv[1]);
            acc[2] += fabsf(fv - muv[2]);
            acc[3] += fabsf(fv - muv[3]);
        }
    }

    // fold positive term and 1/var, sum this thread's 4 samples
    float samp = 0.0f;
    #pragma unroll
    for (int p = 0; p < 4; ++p) {
        const int i = i0 + irow + 4 * p;
        const float meanAbs = acc[p] * (1.0f / NS);
        const float pos     = fabsf(muv[p] - f_r[(size_t)i * DF + d]);
        const float v       = var_ws[base + (size_t)i * DF + d];
        samp += (meanAbs - pos) / v;
    }

    red[t] = samp;
    __syncthreads();
    for (int off = 128; off > 0; off >>= 1) {
        if (t < off) red[t] += red[t + off];
        __syncthreads();
    }
    if (t == 0) atomicAdd(out, red[0] * (1.0f / NS));
}

extern "C" void kernel_launch(void* const* d_in, const int* in_sizes, int n_in,
                              void* d_out, int out_size, void* d_ws, size_t ws_size,
                              hipStream_t stream) {
    (void)in_sizes; (void)n_in; (void)out_size; (void)ws_size;
    const float* fea0  = (const float*)d_in[0];
    const float* fea1  = (const float*)d_in[1];
    const float* fea2  = (const float*)d_in[2];
    const float* fea3  = (const float*)d_in[3];
    const float* W_mu  = (const float*)d_in[4];
    const float* b_mu  = (const float*)d_in[5];
    const float* W_var = (const float*)d_in[6];
    const float* b_var = (const float*)d_in[7];
    float* out = (float*)d_out;

    float* mu_ws  = (float*)d_ws;                 // NL*NS*DF floats = 768 KB
    float* var_ws = mu_ws + (size_t)NL * NS * DF; // another 768 KB

    ib_zero<<<1, 32, 0, stream>>>(out);
    ib_head_wmma<<<192, 128, 0, stream>>>(fea1, fea2, fea3,
                                          W_mu, b_mu, W_var, b_var,
                                          mu_ws, var_ws);
    dim3 grid(NS / TI, NL);
    ib_pairwise<<<grid, 256, 0, stream>>>(fea0, fea1, fea2, mu_ws, var_ws, out);
}